// DenseMPS_25881472925902
// MI455X (gfx1250) — compile-verified
//
#include <hip/hip_runtime.h>
#include <cstdint>
#include <cstddef>

#define N_CORES 128
#define CHI     64
#define VOCAB   256
#define MATSZ   (CHI * CHI)   // 4096 floats = 16 KB

// ---------------------------------------------------------------------------
// CDNA5 async global->LDS copy (ASYNCcnt path).  Builtin signature
// (confirmed by round-1 diagnostic): (v4i AS1*, v4i AS3*, imm offset, cpol).
// The imm offset maps to the instruction IOFFSET, which hardware adds to BOTH
// the global and the LDS address (CDNA5 ISA) — our global
// and LDS tiles share the same layout, so one base pair + 32 immediates.
// ---------------------------------------------------------------------------
#if defined(__HIP_DEVICE_COMPILE__) && defined(__has_builtin)
# if __has_builtin(__builtin_amdgcn_global_load_async_to_lds_b128)
#  define HAVE_ASYNC_BUILTIN 1
# endif
#endif
#ifndef HAVE_ASYNC_BUILTIN
# define HAVE_ASYNC_BUILTIN 0
#endif

typedef int   v4i __attribute__((ext_vector_type(4)));
typedef float v2f __attribute__((ext_vector_type(2)));
typedef __attribute__((address_space(1))) v4i* as1_v4i_p;
typedef __attribute__((address_space(3))) v4i* as3_v4i_p;

template <int OFF>
__device__ __forceinline__ void async_b128_off(const float* g, float* l) {
#if HAVE_ASYNC_BUILTIN
  __builtin_amdgcn_global_load_async_to_lds_b128((as1_v4i_p)g, (as3_v4i_p)l, OFF, 0);
#else
  unsigned           loff = (unsigned)(uintptr_t)l;
  unsigned long long ga   = (unsigned long long)(uintptr_t)g;
  asm volatile("global_load_async_to_lds_b128 %0, %1, off offset:%c2"
               :: "v"(loff), "v"(ga), "i"(OFF) : "memory");
#endif
}

__device__ __forceinline__ void wait_async_zero() {
  // memory clobber: compiler must not hoist LDS reads above this wait
  asm volatile("s_wait_asynccnt 0x0" ::: "memory");
}

// One full 16KB matrix per wave: 32 x b128, lane covers 16B per 512B chunk.
// gbase is wave-uniform (readfirstlane'd token) -> saddr form; chunk offset
// is an instruction immediate -> no per-chunk address math.
__device__ __forceinline__ void prefetch_matrix(const float* gbase, float* sbase, int lane) {
  const float* g = gbase + lane * 4;
  float*       s = sbase + lane * 4;
#define PF4(K) async_b128_off<(K)*512>(g, s);       async_b128_off<((K)+1)*512>(g, s); \
               async_b128_off<((K)+2)*512>(g, s);   async_b128_off<((K)+3)*512>(g, s)
  PF4(0); PF4(4); PF4(8); PF4(12); PF4(16); PF4(20); PF4(24); PF4(28);
#undef PF4
}

// ---------------------------------------------------------------------------
// One wave (32 lanes) per batch row.  Lane owns output columns 2l and 2l+1
// (packed v2f).  Double-buffered 16KB step matrix in LDS, async-prefetched
// one step ahead; compute(i) overlaps DMA(i+1).
// ---------------------------------------------------------------------------
__global__ __launch_bounds__(32)
void mps_chain_kernel(const int*   __restrict__ x,
                      const float* __restrict__ embL,
                      const float* __restrict__ embM,
                      const float* __restrict__ embR,
                      float*       __restrict__ out)
{
  __shared__ __align__(16) float buf[2][MATSZ];   // 32 KB double buffer
  __shared__ __align__(16) float Lv[CHI];         // current left vector
  __shared__ __align__(16) int   toks[N_CORES];   // this row's tokens

  const int row  = blockIdx.x;
  const int lane = threadIdx.x;                   // 0..31, wave32

  // Stage the 128 tokens for this row into LDS (4 per lane, one b128 load).
  ((int4*)toks)[lane] = ((const int4*)(x + (size_t)row * N_CORES))[lane];

  // L0 = emb_left[tok0]  (token made wave-uniform -> scalar base)
  {
    const int tok0 = __builtin_amdgcn_readfirstlane(toks[0]);
    v2f l0 = ((const v2f*)(embL + (size_t)tok0 * CHI))[lane];
    ((v2f*)Lv)[lane] = l0;
  }

  // Prefetch matrix for step 1 into buf[1].
  {
    const int tok = __builtin_amdgcn_readfirstlane(toks[1]);
    prefetch_matrix(embM + ((size_t)0 * VOCAB + tok) * MATSZ, buf[1], lane);
  }

  float ls = 0.0f;

#pragma unroll 1
  for (int i = 1; i <= N_CORES - 2; ++i) {
    const int cur = i & 1;

    wait_async_zero();                       // buf[cur] now complete

    if (i < N_CORES - 2) {                   // kick off step i+1 into the other buffer
      const int tok = __builtin_amdgcn_readfirstlane(toks[i + 1]);
      prefetch_matrix(embM + ((size_t)i * VOCAB + tok) * MATSZ, buf[cur ^ 1], lane);
    }

    // Lnew[d] = sum_c Lv[c] * M[c][d]   (lane -> packed d = {2l, 2l+1})
    v2f acc0 = {0.f, 0.f}, acc1 = {0.f, 0.f}, acc2 = {0.f, 0.f}, acc3 = {0.f, 0.f};
    const float* M = buf[cur];
#pragma unroll
    for (int c4 = 0; c4 < CHI / 4; ++c4) {
      const float4 lc = ((const float4*)Lv)[c4];     // uniform addr -> LDS broadcast
      const v2f*   Mr = (const v2f*)(M + c4 * 4 * CHI);
      v2f m0 = Mr[lane];                             // row c
      v2f m1 = Mr[lane + 32];                        // row c+1 (CHI floats = 32 v2f)
      v2f m2 = Mr[lane + 64];                        // row c+2
      v2f m3 = Mr[lane + 96];                        // row c+3
      v2f l0 = {lc.x, lc.x}, l1 = {lc.y, lc.y}, l2 = {lc.z, lc.z}, l3 = {lc.w, lc.w};
      acc0 = __builtin_elementwise_fma(l0, m0, acc0);   // -> v_pk_fma_f32
      acc1 = __builtin_elementwise_fma(l1, m1, acc1);
      acc2 = __builtin_elementwise_fma(l2, m2, acc2);
      acc3 = __builtin_elementwise_fma(l3, m3, acc3);
    }
    v2f a = (acc0 + acc1) + (acc2 + acc3);

    // Renormalize every PARTIAL_LEN=8 steps (i % 8 == 0).
    if ((i & 7) == 0) {
      float p = fmaf(a.x, a.x, a.y * a.y);
#pragma unroll
      for (int off = 16; off > 0; off >>= 1)
        p += __shfl_xor(p, off, 32);
      const float nrm = sqrtf(p);
      ls += logf(nrm);
      const float inv = 1.0f / nrm;
      a.x *= inv;
      a.y *= inv;
    }

    // Publish L for the next step's broadcasts (same wave: in-order LDS).
    ((v2f*)Lv)[lane] = a;
  }

  // final = dot(L, emb_right[tok_last]);  out = ls + log|final|
  {
    const int tokN = __builtin_amdgcn_readfirstlane(toks[N_CORES - 1]);
    const v2f r  = ((const v2f*)(embR + (size_t)tokN * CHI))[lane];
    const v2f lf = ((const v2f*)Lv)[lane];
    float f = fmaf(lf.x, r.x, lf.y * r.y);
#pragma unroll
    for (int off = 16; off > 0; off >>= 1)
      f += __shfl_xor(f, off, 32);
    if (lane == 0)
      out[row] = ls + logf(fabsf(f));
  }
}

// ---------------------------------------------------------------------------
extern "C" void kernel_launch(void* const* d_in, const int* in_sizes, int n_in,
                              void* d_out, int out_size, void* d_ws, size_t ws_size,
                              hipStream_t stream) {
  (void)in_sizes; (void)n_in; (void)d_ws; (void)ws_size;
  const int*   x    = (const int*)  d_in[0];
  const float* embL = (const float*)d_in[1];
  const float* embM = (const float*)d_in[2];
  const float* embR = (const float*)d_in[3];
  float*       out  = (float*)d_out;

  const int rows = out_size;                 // BATCH (8192)
  mps_chain_kernel<<<dim3(rows), dim3(32), 0, stream>>>(x, embL, embM, embR, out);
}